// DctAtt_4449586118719
// MI455X (gfx1250) — compile-verified
//
#include <hip/hip_runtime.h>
#include <hip/hip_bf16.h>
#include <math.h>

typedef __attribute__((ext_vector_type(2))) float v2f;
typedef __attribute__((ext_vector_type(8))) float v8f;

#define NCOL 4096
#define NROW 32768   // B*D = 32*1024
#define DDIM 1024
#define BDIM 32
#define KPARTS 4
#define KSLICE (NCOL / KPARTS)   // 1024 columns per wave-slice

// ---------------------------------------------------------------------------
// Kernel A: w[n] = sum_k dw_w[k] * C[n,k]   (DCT matrix folded with dw_w)
// ---------------------------------------------------------------------------
__global__ void dct_weight_kernel(const float* __restrict__ dw_w,
                                  float* __restrict__ w) {
    int n = blockIdx.x * blockDim.x + threadIdx.x;
    if (n >= NCOL) return;
    const float PI = 3.14159265358979323846f;
    float base = PI * (2.0f * (float)n + 1.0f) / (2.0f * (float)NCOL);
    float s = sqrtf(2.0f / (float)NCOL);
    float acc = 0.0f;
#pragma unroll
    for (int k = 0; k < 5; ++k) {
        float scale = (k == 0) ? s * 0.70710678118654752f : s;
        acc += dw_w[k] * cosf(base * (float)k) * scale;
    }
    w[n] = acc;
}

// ---------------------------------------------------------------------------
// Kernel B: partial GEMV via V_WMMA_F32_16X16X4_F32, K split 4 ways for
// memory-level parallelism (8192 waves -> ~8/SIMD -> >23MB loads in flight).
// One wave: 16 rows x 1024-column K-slice.
// A-matrix f32 16x4 layout: lane L<16 holds (K=k,k+1) of row L, lane L+16
// holds (K=k+2,k+3) -> one float2 load per lane. Only B column 0 (lanes
// 0/16) feeds the D column we read; other columns' values are discarded.
// ---------------------------------------------------------------------------
__global__ __launch_bounds__(256) void gemv_wmma_kernel(
        const float* __restrict__ x, const float* __restrict__ w,
        float* __restrict__ attp) {
    __shared__ float lw[KSLICE];   // 4 KB slice of w
    int tid   = threadIdx.x;
    int part  = blockIdx.x & (KPARTS - 1);
    int rowBlk = blockIdx.x >> 2;
    int kBase = part * KSLICE;

    for (int i = tid; i < KSLICE; i += 256) lw[i] = w[kBase + i];
    __syncthreads();

    int lane    = tid & 31;
    int waveId  = tid >> 5;
    int rowBase = (rowBlk * 8 + waveId) * 16;
    int off2    = ((lane >> 4) & 1) * 2;   // 0 for lanes 0-15, 2 for lanes 16-31

    const float* xrow = x + (size_t)(rowBase + (lane & 15)) * NCOL + kBase + off2;
    const float* wl   = lw + off2;

    v8f c0 = {};
    v8f c1 = {};

    for (int k = 0; k < KSLICE; k += 64) {
        __builtin_prefetch(xrow + k + 512, 0, 3);   // global_prefetch_b8
#pragma unroll
        for (int j = 0; j < 64; j += 8) {
            v2f a0 = *(const v2f*)(xrow + k + j);
            v2f b0 = *(const v2f*)(wl   + k + j);
            c0 = __builtin_amdgcn_wmma_f32_16x16x4_f32(
                     false, a0, false, b0, (short)0, c0, false, false);
            v2f a1 = *(const v2f*)(xrow + k + j + 4);
            v2f b1 = *(const v2f*)(wl   + k + j + 4);
            c1 = __builtin_amdgcn_wmma_f32_16x16x4_f32(
                     false, a1, false, b1, (short)0, c1, false, false);
        }
    }

    v8f c = c0 + c1;
    // D column 0: lane 0 holds M=0..7 in c[0..7], lane 16 holds M=8..15.
    if ((lane & 15) == 0) {
        int m0 = part * NROW + rowBase + ((lane >> 4) << 3);
#pragma unroll
        for (int i = 0; i < 8; ++i) attp[m0 + i] = c[i];
    }
}

// ---------------------------------------------------------------------------
// Kernel C: fold 4 K-partials + bias into att, and compute mean / inv-std.
// ---------------------------------------------------------------------------
__global__ __launch_bounds__(1024) void stats_kernel(
        const float* __restrict__ attp, const float* __restrict__ dw_b,
        float* __restrict__ att, float* __restrict__ stats) {
    __shared__ float ssum[1024];
    __shared__ float ssq[1024];
    int tid = threadIdx.x;
    float bias = dw_b[0];
    float s = 0.0f, q = 0.0f;
    for (int i = tid; i < NROW; i += 1024) {
        float v = attp[i] + attp[NROW + i] + attp[2 * NROW + i]
                + attp[3 * NROW + i] + bias;
        att[i] = v;
        s += v; q += v * v;
    }
    ssum[tid] = s; ssq[tid] = q;
    __syncthreads();
    for (int off = 512; off > 0; off >>= 1) {
        if (tid < off) { ssum[tid] += ssum[tid + off]; ssq[tid] += ssq[tid + off]; }
        __syncthreads();
    }
    if (tid == 0) {
        float mean = ssum[0] * (1.0f / (float)NROW);
        float var  = ssq[0] * (1.0f / (float)NROW) - mean * mean;
        stats[0] = mean;
        stats[1] = rsqrtf(var + 1e-5f);
    }
}

// ---------------------------------------------------------------------------
// Kernel D: BN -> exact GELU -> affine -> softmax over D (1024) per batch row.
// Writes out[0:32768] = p,  out[32768:65536] = 1 - p.
// ---------------------------------------------------------------------------
__global__ __launch_bounds__(1024) void softmax_kernel(
        const float* __restrict__ att, const float* __restrict__ stats,
        const float* __restrict__ gamma, const float* __restrict__ beta,
        const float* __restrict__ conv_w, const float* __restrict__ conv_b,
        float* __restrict__ out) {
    __shared__ float red[1024];
    int b = blockIdx.x, d = threadIdx.x;

    float mean = stats[0], inv = stats[1];
    float a  = att[b * DDIM + d];
    float an = (a - mean) * inv * gamma[0] + beta[0];
    float g  = 0.5f * an * (1.0f + erff(an * 0.70710678118654752f));  // exact gelu
    float y  = g * conv_w[0] + conv_b[0];

    red[d] = y;
    __syncthreads();
    for (int off = 512; off > 0; off >>= 1) {
        if (d < off) red[d] = fmaxf(red[d], red[d + off]);
        __syncthreads();
    }
    float m = red[0];
    __syncthreads();

    float e = expf(y - m);
    red[d] = e;
    __syncthreads();
    for (int off = 512; off > 0; off >>= 1) {
        if (d < off) red[d] += red[d + off];
        __syncthreads();
    }
    float p = e / red[0];

    out[b * DDIM + d]        = p;
    out[NROW + b * DDIM + d] = 1.0f - p;
}

// ---------------------------------------------------------------------------
extern "C" void kernel_launch(void* const* d_in, const int* in_sizes, int n_in,
                              void* d_out, int out_size, void* d_ws, size_t ws_size,
                              hipStream_t stream) {
    const float* x      = (const float*)d_in[0];  // (32,1024,4096)
    const float* dw_w   = (const float*)d_in[1];  // (5,)
    const float* dw_b   = (const float*)d_in[2];  // (1,)
    const float* gamma  = (const float*)d_in[3];  // (1,)
    const float* beta   = (const float*)d_in[4];  // (1,)
    const float* conv_w = (const float*)d_in[5];  // (1,)
    const float* conv_b = (const float*)d_in[6];  // (1,)
    float* out = (float*)d_out;                   // 65536 floats

    float* ws    = (float*)d_ws;
    float* w     = ws;                               // 4096
    float* attp  = ws + NCOL;                        // 4 * 32768 partials
    float* att   = attp + KPARTS * NROW;             // 32768
    float* stats = att + NROW;                       // 2

    dct_weight_kernel<<<NCOL / 256, 256, 0, stream>>>(dw_w, w);
    gemv_wmma_kernel<<<(NROW / 128) * KPARTS, 256, 0, stream>>>(x, w, attp);
    stats_kernel<<<1, 1024, 0, stream>>>(attp, dw_b, att, stats);
    softmax_kernel<<<BDIM, 1024, 0, stream>>>(att, stats, gamma, beta,
                                              conv_w, conv_b, out);
}